// StemLayer_14937896256224
// MI455X (gfx1250) — compile-verified
//
#include <hip/hip_runtime.h>
#include <math.h>

typedef __attribute__((ext_vector_type(2))) float v2f;
typedef __attribute__((ext_vector_type(4))) float v4f;
typedef __attribute__((ext_vector_type(8))) float v8f;

#define K_TAPS 27
#define CIN    4
#define COUT   64

// ---------------------------------------------------------------------------
// Kernel 1: gathered GEMM via V_WMMA_F32_16X16X4_F32 + per-channel sum/sumsq.
// One wave32 = one 16-node x 64-channel tile per loop trip.
//   A (16x4, f32): lane l<16  -> node (base+l), channels {0,1}
//                  lane l>=16 -> node (base+l-16), channels {2,3}
//   B (4x16, f32): VGPR0 rows {K0 | K2}, VGPR1 rows {K1 | K3} (lane halves)
//   C/D (16x16):   VGPR j, lane l -> y[base + j + 8*(l>>4)][t*16 + (l&15)]
// Tap loop unrolled 9x (not 27x) to stay under 256 VGPRs: avoids
// s_set_vgpr_msb overhead and keeps occupancy up for the random gathers.
// ---------------------------------------------------------------------------
__global__ __launch_bounds__(256) void conv_stats_kernel(
    const float* __restrict__ x,      // [N_in][4]
    const float* __restrict__ w,      // [27][4][64]
    const int*   __restrict__ nidx,   // [N_out][27]
    float*       __restrict__ y,      // [N_out][64] (workspace)
    float*       __restrict__ stats,  // [0:64)=sum  [64:128)=sumsq
    int n_out)
{
    // swizzled weights: s_w[k][t][lane] = (w[k][2*ch][t*16+m], w[k][2*ch+1][t*16+m])
    __shared__ v2f   s_w[K_TAPS * 4 * 32];
    __shared__ float s_stat[128];

    const int tid = threadIdx.x;
    for (int i = tid; i < K_TAPS * 4 * 32; i += 256) {
        int lane = i & 31;
        int t    = (i >> 5) & 3;
        int k    = i >> 7;
        int m    = lane & 15;
        int ch   = lane >> 4;
        int o    = t * 16 + m;
        v2f val;
        val.x = w[(k * CIN + 2 * ch + 0) * COUT + o];
        val.y = w[(k * CIN + 2 * ch + 1) * COUT + o];
        s_w[i] = val;
    }
    if (tid < 128) s_stat[tid] = 0.0f;
    __syncthreads();

    const int lane   = tid & 31;
    const int m      = lane & 15;
    const int ch     = lane >> 4;                 // 0 -> K{0,1}, 1 -> K{2,3}
    const int wave   = blockIdx.x * 8 + (tid >> 5);
    const int nwaves = gridDim.x * 8;
    const int ntiles = (n_out + 15) >> 4;

    float sum[4] = {0.f, 0.f, 0.f, 0.f};
    float sq[4]  = {0.f, 0.f, 0.f, 0.f};

    for (int tile = wave; tile < ntiles; tile += nwaves) {
        const int base = tile << 4;
        int row  = base + m;
        int grow = row < n_out ? row : (n_out - 1);   // clamp gather (EXEC stays full)
        const int* __restrict__ nrow = nidx + (long)grow * K_TAPS;

        v8f acc0 = {}, acc1 = {}, acc2 = {}, acc3 = {};
        #pragma unroll 9
        for (int k = 0; k < K_TAPS; ++k) {
            unsigned idx = (unsigned)nrow[k];         // non-negative: zero-extend
            v2f a = *(const v2f*)(x + (unsigned long)idx * CIN + ch * 2);
            const v2f* bw = &s_w[(k * 4) * 32 + lane];
            v2f b0 = bw[0 * 32];
            v2f b1 = bw[1 * 32];
            v2f b2 = bw[2 * 32];
            v2f b3 = bw[3 * 32];
            acc0 = __builtin_amdgcn_wmma_f32_16x16x4_f32(false, a, false, b0, (short)0, acc0, false, false);
            acc1 = __builtin_amdgcn_wmma_f32_16x16x4_f32(false, a, false, b1, (short)0, acc1, false, false);
            acc2 = __builtin_amdgcn_wmma_f32_16x16x4_f32(false, a, false, b2, (short)0, acc2, false, false);
            acc3 = __builtin_amdgcn_wmma_f32_16x16x4_f32(false, a, false, b3, (short)0, acc3, false, false);
        }

        #pragma unroll
        for (int j = 0; j < 8; ++j) {
            int r = base + j + 8 * ch;
            if (r < n_out) {
                float e0 = acc0[j], e1 = acc1[j], e2 = acc2[j], e3 = acc3[j];
                float* yp = y + (long)r * COUT + m;
                yp[0]  = e0;
                yp[16] = e1;
                yp[32] = e2;
                yp[48] = e3;
                sum[0] += e0; sq[0] += e0 * e0;
                sum[1] += e1; sq[1] += e1 * e1;
                sum[2] += e2; sq[2] += e2 * e2;
                sum[3] += e3; sq[3] += e3 * e3;
            }
        }
    }

    #pragma unroll
    for (int t = 0; t < 4; ++t) {
        atomicAdd(&s_stat[t * 16 + m],      sum[t]);
        atomicAdd(&s_stat[64 + t * 16 + m], sq[t]);
    }
    __syncthreads();
    if (tid < 128) atomicAdd(&stats[tid], s_stat[tid]);
}

// ---------------------------------------------------------------------------
// Kernel 2: fold BN (training-mode batch stats) + gamma/beta into per-channel
// scale/bias: scale = gamma * rsqrt(var + eps), bias = beta - mean * scale.
// ---------------------------------------------------------------------------
__global__ void bn_finalize_kernel(const float* __restrict__ gamma,
                                   const float* __restrict__ beta,
                                   float* __restrict__ stats, float inv_n)
{
    int o = threadIdx.x;
    if (o < COUT) {
        float mean = stats[o] * inv_n;
        float var  = stats[64 + o] * inv_n - mean * mean;
        float rs   = rsqrtf(var + 1e-5f);
        float sc   = gamma[o] * rs;
        stats[128 + o] = sc;
        stats[192 + o] = beta[o] - mean * sc;
    }
}

// ---------------------------------------------------------------------------
// Kernel 3: max-pool over 8 children. ReLU is monotone, so
// max_j relu(a_j) == relu(max_j a_j): one ReLU after the max.
// Thread handles one parent x 4 channels (float4 path).
// ---------------------------------------------------------------------------
__global__ __launch_bounds__(256) void pool_kernel(
    const float* __restrict__ y,      // [N_out][64]
    const int*   __restrict__ pidx,   // [N_pool][8]
    const float* __restrict__ stats,  // scale at 128, bias at 192
    const int*   __restrict__ depth,
    float*       __restrict__ out,
    int n_pool, int out_size)
{
    int gid = blockIdx.x * blockDim.x + threadIdx.x;
    if (gid == 0 && out_size > n_pool * COUT) {
        // second tuple element: depth - 2 (integer)
        ((int*)out)[out_size - 1] = depth[0] - 2;
    }
    if (gid >= n_pool * 16) return;

    int p  = gid >> 4;
    int c4 = (gid & 15) * 4;

    v4f sc = *(const v4f*)(stats + 128 + c4);
    v4f bi = *(const v4f*)(stats + 192 + c4);

    float b0 = -3.402823466e38f, b1 = b0, b2 = b0, b3 = b0;
    #pragma unroll
    for (int j = 0; j < 8; ++j) {
        unsigned idx = (unsigned)pidx[p * 8 + j];
        v4f v = *(const v4f*)(y + (unsigned long)idx * COUT + c4);
        b0 = fmaxf(b0, v.x * sc.x + bi.x);
        b1 = fmaxf(b1, v.y * sc.y + bi.y);
        b2 = fmaxf(b2, v.z * sc.z + bi.z);
        b3 = fmaxf(b3, v.w * sc.w + bi.w);
    }
    v4f r;
    r.x = fmaxf(b0, 0.0f);
    r.y = fmaxf(b1, 0.0f);
    r.z = fmaxf(b2, 0.0f);
    r.w = fmaxf(b3, 0.0f);
    *(v4f*)(out + (long)p * COUT + c4) = r;
}

// ---------------------------------------------------------------------------
extern "C" void kernel_launch(void* const* d_in, const int* in_sizes, int n_in,
                              void* d_out, int out_size, void* d_ws, size_t ws_size,
                              hipStream_t stream)
{
    const float* x     = (const float*)d_in[0];
    const float* w     = (const float*)d_in[1];
    const float* gamma = (const float*)d_in[2];
    const float* beta  = (const float*)d_in[3];
    const int*   nidx  = (const int*)d_in[4];
    const int*   pidx  = (const int*)d_in[5];
    const int*   depth = (const int*)d_in[6];

    const int n_out  = in_sizes[4] / K_TAPS;   // 400000
    const int n_pool = in_sizes[5] / 8;        // 60000

    float* stats = (float*)d_ws;               // 256 floats: sum, sumsq, scale, bias
    float* y     = (float*)d_ws + 256;         // [n_out][64]

    hipMemsetAsync(stats, 0, 128 * sizeof(float), stream);

    conv_stats_kernel<<<512, 256, 0, stream>>>(x, w, nidx, y, stats, n_out);
    bn_finalize_kernel<<<1, 64, 0, stream>>>(gamma, beta, stats, 1.0f / (float)n_out);

    const int pool_threads = n_pool * 16;
    pool_kernel<<<(pool_threads + 255) / 256, 256, 0, stream>>>(
        y, pidx, stats, depth, (float*)d_out, n_pool, out_size);
}